// DenoiseEncoder_80693845557942
// MI455X (gfx1250) — compile-verified
//
#include <hip/hip_runtime.h>
#include <hip/hip_bf16.h>

// ---------------------------------------------------------------------------
// LightGCN-style 2-layer propagation over 3 weighted relations.
//   x1 = sum_r (w_r/3) * scatter_add(emb[dst_r] -> src_r)
//   x2 = sum_r (w_r/3) * scatter_add(x1[dst_r]  -> src_r)
//   out = (emb + x1 + x2) / 3        (all 150000 x 128, fp32)
//
// MI455X strategy:
//  * The whole embedding working set (2 x 76.8MB) fits in the 192MB L2;
//    bottleneck is L2 fp32 atomic throughput, not HBM.
//  * Edge index streams (96MB over both layers) are read with non-temporal
//    loads so they do not evict the L2-resident embeddings; prefetched ahead.
//  * wave32: one wave per edge; lane owns a float4 slice -> 512B coalesced
//    gather + 128 coalesced no-return global_atomic_add_f32 per edge.
//  * Final mean runs on the matrix pipe: per 16x16 tile,
//    D = (1/3 * I) x (x0+x1+x2) via 4 chained v_wmma_f32_16x16x4_f32.
// ---------------------------------------------------------------------------

typedef float v2f __attribute__((ext_vector_type(2)));
typedef float v8f __attribute__((ext_vector_type(8)));

#define TOTAL_NODES 150000
#define EMBED_DIM   128
#define LEVEL_FLOATS ((size_t)TOTAL_NODES * EMBED_DIM)   // 19.2M floats

// ---- zero the two workspace levels (float4 stores, grid-stride) -----------
__global__ void zero_ws_kernel(float4* __restrict__ p, size_t n4) {
    size_t i = (size_t)blockIdx.x * blockDim.x + threadIdx.x;
    const size_t stride = (size_t)gridDim.x * blockDim.x;
    const float4 z = make_float4(0.f, 0.f, 0.f, 0.f);
    for (; i < n4; i += stride) p[i] = z;
}

// ---- one wave32 per edge: gather row x[dst], atomically add w*row to agg[src]
__global__ void scatter_rel_kernel(const float* __restrict__ x,
                                   float* __restrict__ agg,
                                   const int* __restrict__ src,
                                   const int* __restrict__ dst,
                                   float w, int num_edges) {
    const int lane = threadIdx.x & 31;
    const int wave = threadIdx.x >> 5;
    const int wpb  = blockDim.x >> 5;
    const int estride = gridDim.x * wpb;

    for (int e = blockIdx.x * wpb + wave; e < num_edges; e += estride) {
        // Non-temporal: index stream is read-once; keep L2 for embeddings.
        int s = __builtin_amdgcn_readfirstlane(__builtin_nontemporal_load(src + e));
        int d = __builtin_amdgcn_readfirstlane(__builtin_nontemporal_load(dst + e));
        // Prefetch next iteration's indices (global_prefetch_b8).
        __builtin_prefetch(src + e + estride, 0, 0);
        __builtin_prefetch(dst + e + estride, 0, 0);

        const float4 v = *(const float4*)(x + (size_t)d * EMBED_DIM + lane * 4);
        float* p = agg + (size_t)s * EMBED_DIM + lane * 4;
        atomicAdd(p + 0, w * v.x);   // global_atomic_add_f32, no-return
        atomicAdd(p + 1, w * v.y);
        atomicAdd(p + 2, w * v.z);
        atomicAdd(p + 3, w * v.w);
    }
}

// ---- fused mean epilogue on the matrix pipe -------------------------------
// One wave per 16x16 tile: out_tile = (1/3 * I16) x (x0+x1+x2)_tile,
// K-chunked into 4 x V_WMMA_F32_16X16X4_F32 accumulating in C.
// A (16x4, ISA layout): lane l(0..15)/hi half: V0 = A(l, 2*hi), V1 = A(l, 2*hi+1)
// B (4x16, mirrored):   V0 = B(2*hi, l),      V1 = B(2*hi+1, l)
__global__ void mean_wmma_kernel(const float* __restrict__ x0,
                                 const float* __restrict__ x1,
                                 const float* __restrict__ x2,
                                 float* __restrict__ out) {
    const int lane  = threadIdx.x & 31;
    const int wave  = threadIdx.x >> 5;
    const int tile  = blockIdx.x * (blockDim.x >> 5) + wave;  // 0 .. 74999
    const int tileM = tile >> 3;   // node tile: 0 .. 9374  (150000/16)
    const int tileN = tile & 7;    // dim  tile: 0 .. 7     (128/16)
    const int l  = lane & 15;
    const int hi = lane >> 4;      // 0 or 1

    const size_t rowBase = (size_t)tileM * 16;
    const int    colBase = tileN * 16;
    const float  third   = 1.0f / 3.0f;

    v8f acc = {0.f, 0.f, 0.f, 0.f, 0.f, 0.f, 0.f, 0.f};
#pragma unroll
    for (int c = 0; c < 4; ++c) {
        const int n0 = 4 * c + 2 * hi;   // node row feeding B VGPR0 (K = n0)
        const int n1 = n0 + 1;           // node row feeding B VGPR1 (K = n1)
        const size_t off0 = (rowBase + n0) * EMBED_DIM + colBase + l;
        const size_t off1 = (rowBase + n1) * EMBED_DIM + colBase + l;
        v2f B;
        B.x = x0[off0] + x1[off0] + x2[off0];
        B.y = x0[off1] + x1[off1] + x2[off1];
        v2f A;                            // columns of (1/3)*I16, chunk c
        A.x = (l == n0) ? third : 0.0f;
        A.y = (l == n1) ? third : 0.0f;
        acc = __builtin_amdgcn_wmma_f32_16x16x4_f32(
            /*neg_a=*/false, A, /*neg_b=*/false, B,
            /*c_mod=*/(short)0, acc, /*reuse_a=*/false, /*reuse_b=*/false);
    }
    // C/D layout: lane l (hi half): VGPR v holds (M = v + 8*hi, N = l)
#pragma unroll
    for (int v = 0; v < 8; ++v) {
        out[(rowBase + v + 8 * hi) * EMBED_DIM + colBase + l] = acc[v];
    }
}

extern "C" void kernel_launch(void* const* d_in, const int* in_sizes, int n_in,
                              void* d_out, int out_size, void* d_ws, size_t ws_size,
                              hipStream_t stream) {
    const float* emb = (const float*)d_in[0];
    const int* src_r[3] = { (const int*)d_in[1], (const int*)d_in[3], (const int*)d_in[5] };
    const int* dst_r[3] = { (const int*)d_in[2], (const int*)d_in[4], (const int*)d_in[6] };
    const int  num_edges = in_sizes[1];
    float* out = (float*)d_out;

    // Workspace: two fp32 levels, 76.8MB each (ws_size must be >= 153.6MB).
    float* x1 = (float*)d_ws;
    float* x2 = x1 + LEVEL_FLOATS;

    // Relation weights folded with the per-layer 1/3 normalization.
    const float w[3] = { 1.0f / 3.0f, 0.5f / 3.0f, 2.0f / 3.0f };

    // 1) zero both accumulator levels (x2 untouched until layer 2).
    {
        const size_t n4 = (2 * LEVEL_FLOATS) / 4;
        zero_ws_kernel<<<2048, 256, 0, stream>>>((float4*)d_ws, n4);
    }
    // 2) layer 1: emb -> x1
    for (int r = 0; r < 3; ++r)
        scatter_rel_kernel<<<4096, 256, 0, stream>>>(emb, x1, src_r[r], dst_r[r],
                                                     w[r], num_edges);
    // 3) layer 2: x1 -> x2
    for (int r = 0; r < 3; ++r)
        scatter_rel_kernel<<<4096, 256, 0, stream>>>(x1, x2, src_r[r], dst_r[r],
                                                     w[r], num_edges);
    // 4) out = (emb + x1 + x2)/3 on the WMMA pipe.
    //    75000 tiles, 8 waves/block -> 9375 blocks, all waves full (EXEC all-1s).
    mean_wmma_kernel<<<9375, 256, 0, stream>>>(emb, x1, x2, out);
}